// bin_feature_31963146617325
// MI455X (gfx1250) — compile-verified
//
#include <hip/hip_runtime.h>

// ---------------------------------------------------------------------------
// bin-feature encoder + Linear + ReLU, collapsed analytically:
//   y[b,f,d] = relu( P[f0][d] + frac*W^T[f0][d] + bias[d] )
// where P is the exclusive prefix sum of W over the bin axis.
//
// Phase 1 (prep, tiny: 1 MB of W): build Pt[2000][128] (exclusive prefix) and
//   Wt[2000][128] (transpose) in workspace. The 16-wide blocked scan is done
//   as S = L_strict x T using chained V_WMMA_F32_16X16X4_F32 (exact fp32).
// Phase 2 (hot, ~34 MB HBM): per (b,f) gather the two 512B rows (L2-resident)
//   and emit the fused fma+relu. Bandwidth-bound: ~1.5 us at 23.3 TB/s.
// ---------------------------------------------------------------------------

typedef __attribute__((ext_vector_type(2))) float v2f;
typedef __attribute__((ext_vector_type(8))) float v8f;

#define NBINS   2000
#define DDIM    128
#define KBLOCKS (NBINS / 16)   // 125
#define BTOT    (128 * 512)    // B * N_FEATURE waves of work

// --- Phase 1: blocked exclusive scan of W[d][k] along k, via f32 WMMA -------
// Grid: 8 blocks (one per 16-wide d slice) x 32 threads (one wave).
// Wave-uniform serial carry across 125 k-blocks; intra-block exclusive prefix
// computed as L_strict(16x16) x T(16x16) with four 16x16x4 fp32 WMMAs.
// Tile kept entirely in float4 registers (no arrays -> no LDS demotion).
__global__ __launch_bounds__(32)
void prep_scan_wmma(const float* __restrict__ W,
                    float* __restrict__ Pt,     // [NBINS][DDIM] exclusive prefix
                    float* __restrict__ Wt)     // [NBINS][DDIM] transpose of W
{
    const int  lane  = threadIdx.x & 31;
    const int  n     = lane & 15;          // d-column within block; also A-row M
    const bool hi    = lane >= 16;
    const int  dBase = blockIdx.x * 16;
    const float* wRow = W + (size_t)(dBase + n) * NBINS;   // row of W (contiguous k)

    // Constant A operand: strictly-lower-triangular ones, L[m][k] = (k < m).
    // A 16x4 f32 layout: lanes 0-15 hold K={0,1}, lanes 16-31 hold K={2,3}.
    const int kOff = hi ? 2 : 0;
    v2f a0, a1, a2, a3;
    a0.x = ((0  + kOff)     < n) ? 1.0f : 0.0f;  a0.y = ((0  + kOff + 1) < n) ? 1.0f : 0.0f;
    a1.x = ((4  + kOff)     < n) ? 1.0f : 0.0f;  a1.y = ((4  + kOff + 1) < n) ? 1.0f : 0.0f;
    a2.x = ((8  + kOff)     < n) ? 1.0f : 0.0f;  a2.y = ((8  + kOff + 1) < n) ? 1.0f : 0.0f;
    a3.x = ((12 + kOff)     < n) ? 1.0f : 0.0f;  a3.y = ((12 + kOff + 1) < n) ? 1.0f : 0.0f;

    float carry = 0.0f;   // running column sum for this lane's d
    for (int kblk = 0; kblk < KBLOCKS; ++kblk) {
        const int kBase = kblk * 16;

        // q0..q3 = T[0..15][n] = W[dBase+n][kBase+0..15]; 64B-aligned contiguous.
        const float4 q0 = *(const float4*)(wRow + kBase + 0);
        const float4 q1 = *(const float4*)(wRow + kBase + 4);
        const float4 q2 = *(const float4*)(wRow + kBase + 8);
        const float4 q3 = *(const float4*)(wRow + kBase + 12);

        // B operand 4x16 f32: lanes 0-15 rows K={s,s+1}, lanes 16-31 rows K={s+2,s+3}.
        v2f b0, b1, b2, b3;
        b0.x = hi ? q0.z : q0.x;   b0.y = hi ? q0.w : q0.y;
        b1.x = hi ? q1.z : q1.x;   b1.y = hi ? q1.w : q1.y;
        b2.x = hi ? q2.z : q2.x;   b2.y = hi ? q2.w : q2.y;
        b3.x = hi ? q3.z : q3.x;   b3.y = hi ? q3.w : q3.y;

        // Accumulator seeded with the carry (same d-column for all 8 rows held
        // by this lane, so a scalar broadcast is exact).
        v8f c;
#pragma unroll
        for (int i = 0; i < 8; ++i) c[i] = carry;

        // S = carry + L_strict x T  (exact fp32, RNE)
        c = __builtin_amdgcn_wmma_f32_16x16x4_f32(false, a0, false, b0, (short)0, c, false, false);
        c = __builtin_amdgcn_wmma_f32_16x16x4_f32(false, a1, false, b1, (short)0, c, false, false);
        c = __builtin_amdgcn_wmma_f32_16x16x4_f32(false, a2, false, b2, (short)0, c, false, false);
        c = __builtin_amdgcn_wmma_f32_16x16x4_f32(false, a3, false, b3, (short)0, c, false, false);

        // C/D 16x16 layout: VGPR v holds row M=v (lanes 0-15) / M=v+8 (lanes 16-31).
        const int mOff = hi ? 8 : 0;
        float* pRow = Pt + (size_t)(kBase + mOff) * DDIM + dBase + n;
#pragma unroll
        for (int v = 0; v < 8; ++v) pRow[(size_t)v * DDIM] = c[v];

        // Transpose store of W: low half-wave writes kk 0..7 (q0,q1), high
        // half writes kk 8..15 (q2,q3). Constant component selects only.
        float* wtRow = Wt + (size_t)(kBase + mOff) * DDIM + dBase + n;
        wtRow[0 * DDIM] = hi ? q2.x : q0.x;
        wtRow[1 * DDIM] = hi ? q2.y : q0.y;
        wtRow[2 * DDIM] = hi ? q2.z : q0.z;
        wtRow[3 * DDIM] = hi ? q2.w : q0.w;
        wtRow[4 * DDIM] = hi ? q3.x : q1.x;
        wtRow[5 * DDIM] = hi ? q3.y : q1.y;
        wtRow[6 * DDIM] = hi ? q3.z : q1.z;
        wtRow[7 * DDIM] = hi ? q3.w : q1.w;

        // carry += full block column sum = S_excl[15][n] + T[15][n].
        // S_excl[15][n] sits in c[7] of lane 16+n; T[15][n] = q3.w (both halves).
        const float s15 = __shfl(c[7], 16 + n, 32);
        carry = s15 + q3.w;
    }
}

// --- Phase 2: fused gather + fma + relu -------------------------------------
// One wave per (b,f); 32 lanes x float4 = 128 outputs, fully coalesced.
__global__ __launch_bounds__(256)
void gather_fuse(const float* __restrict__ x,
                 const float* __restrict__ Pt,
                 const float* __restrict__ Wt,
                 const float* __restrict__ bias,
                 float* __restrict__ out)
{
    const int wave = (blockIdx.x * 256 + threadIdx.x) >> 5;   // 0 .. BTOT-1
    const int lane = threadIdx.x & 31;
    if (wave >= BTOT) return;

    // Replicate the reference arithmetic exactly: pos = x+1000, f = floor, frac.
    const float pos  = x[wave] + 1000.0f;
    const float f    = floorf(pos);
    const float frac = pos - f;
    int f0 = (int)f;
    f0 = f0 < 0 ? 0 : (f0 > NBINS - 1 ? NBINS - 1 : f0);

    const float4 p = ((const float4*)(Pt + (size_t)f0 * DDIM))[lane];
    const float4 w = ((const float4*)(Wt + (size_t)f0 * DDIM))[lane];
    const float4 b = ((const float4*)bias)[lane];

    float4 r;
    r.x = fmaxf(fmaf(frac, w.x, p.x) + b.x, 0.0f);
    r.y = fmaxf(fmaf(frac, w.y, p.y) + b.y, 0.0f);
    r.z = fmaxf(fmaf(frac, w.z, p.z) + b.z, 0.0f);
    r.w = fmaxf(fmaf(frac, w.w, p.w) + b.w, 0.0f);

    ((float4*)out)[(size_t)wave * 32 + lane] = r;
}

extern "C" void kernel_launch(void* const* d_in, const int* in_sizes, int n_in,
                              void* d_out, int out_size, void* d_ws, size_t ws_size,
                              hipStream_t stream) {
    const float* x    = (const float*)d_in[0];   // [128, 512]
    const float* W    = (const float*)d_in[1];   // [128, 2000] row-major
    const float* bias = (const float*)d_in[2];   // [128]
    float* out = (float*)d_out;                  // [128, 512, 128]

    float* Pt = (float*)d_ws;                    // NBINS*DDIM floats
    float* Wt = Pt + (size_t)NBINS * DDIM;       // NBINS*DDIM floats (2.05 MB total)

    prep_scan_wmma<<<DDIM / 16, 32, 0, stream>>>(W, Pt, Wt);
    gather_fuse<<<BTOT / 8, 256, 0, stream>>>(x, Pt, Wt, bias, out);
}